// FeedforwardController_24799141167316
// MI455X (gfx1250) — compile-verified
//
#include <hip/hip_runtime.h>
#include <math.h>

// ---------------------------------------------------------------------------
// NTM feedforward-controller step for MI455X (gfx1250, wave32).
// Bandwidth-bound: out1.w is 269MB (read-once, > 192MB L2) -> ~11.6us floor
// @ 23.3 TB/s. Big GEMV runs on V_WMMA_F32_16X16X4_F32 with NT loads; the
// K-loop is strength-reduced to pointer bumps so VALU doesn't gate the stream.
// ---------------------------------------------------------------------------

typedef __attribute__((ext_vector_type(2))) float v2f;
typedef __attribute__((ext_vector_type(8))) float v8f;

#define N_MEM   2048
#define WCOL    128
#define DATA    512
#define OUTD    512
#define PPD     64
#define AUGN    262656      // DATA + N_MEM*WCOL = 512*513
#define KCHUNK  512
#define NKCHUNK 513         // 513 * 512 = 262656

// workspace layout (float offsets)
#define OFF_PP      128
#define OFF_RKEY    192
#define OFF_WKEY    320
#define OFF_SCAL    448   // [0]rbeta [1]rgate [2]rgamma [3]wbeta [4]wgate [5]wgamma [6]knr [7]knw
#define OFF_ERASE   512
#define OFF_ADD     640
#define OFF_RSHIFT  768
#define OFF_WSHIFT  2816
#define OFF_RNUM    4864
#define OFF_WNUM    6912
#define OFF_ROWDEN  8960
#define OFF_RGW     11008
#define OFF_WGW     13056
#define OFF_RCONV   15104
#define OFF_WCONV   17152
#define OFF_RFINAL  19200
#define OFF_WFINAL  21248
#define OFF_AUG     23296
#define OFF_PARTIAL 285952  // 23296 + 262656 ; needs 513*256 = 131328 floats
// total ws: 417280 floats = 1.67 MB

struct Ptrs {
    const float *x, *mem, *prevr, *prevw;
    const float *pp1w,*pp1b,*pp2w,*pp2b;
    const float *rkw,*rkb,*rbw,*rbb,*rgw,*rgb,*rsw,*rsb,*rmw,*rmb;
    const float *wkw,*wkb,*wbw,*wbb,*wgw,*wgb,*wsw,*wsb,*wmw,*wmb;
    const float *erw,*erb,*adw,*adb;
    const float *o1w,*o1b,*o2w,*o2b,*o3w,*o3b;
};

__device__ __forceinline__ float lrelu(float v){ return v > 0.f ? v : 0.01f*v; }
__device__ __forceinline__ float sigm(float v){ return 1.f/(1.f+expf(-v)); }

// ---- controller MLP: pp = lrelu(pp2(lrelu(pp1(x)))) : 512->128->64 ---------
__global__ void k_pp(Ptrs p, float* ws) {
    __shared__ float h1[128];
    int t = threadIdx.x;                       // 128 threads
    float a = p.pp1b[t];
    for (int i = 0; i < DATA; ++i) a += p.x[i]*p.pp1w[i*128+t];
    h1[t] = lrelu(a);
    __syncthreads();
    if (t < PPD) {
        float a2 = p.pp2b[t];
        for (int i = 0; i < 128; ++i) a2 += h1[i]*p.pp2w[i*PPD+t];
        ws[OFF_PP+t] = lrelu(a2);
    }
}

// ---- small head projections: keys / erase / add / scalars -----------------
__global__ void k_heads_small(Ptrs p, float* ws) {
    __shared__ float pps[PPD];
    __shared__ float rk[WCOL], wk[WCOL];
    int t = threadIdx.x;                       // 256 threads
    if (t < PPD) pps[t] = ws[OFF_PP+t];
    __syncthreads();
    if (t < WCOL) {
        float a = p.rkb[t];
        for (int i=0;i<PPD;++i) a += pps[i]*p.rkw[i*WCOL+t];
        rk[t] = lrelu(a); ws[OFF_RKEY+t] = rk[t];
        float e = p.erb[t];
        for (int i=0;i<PPD;++i) e += pps[i]*p.erw[i*WCOL+t];
        ws[OFF_ERASE+t] = sigm(e);
    } else {
        int u = t - WCOL;
        float a = p.wkb[u];
        for (int i=0;i<PPD;++i) a += pps[i]*p.wkw[i*WCOL+u];
        wk[u] = lrelu(a); ws[OFF_WKEY+u] = wk[u];
        float d = p.adb[u];
        for (int i=0;i<PPD;++i) d += pps[i]*p.adw[i*WCOL+u];
        ws[OFF_ADD+u] = sigm(d);
    }
    __syncthreads();
    if (t == 0) { float s=0; for(int i=0;i<WCOL;++i) s+=rk[i]*rk[i]; ws[OFF_SCAL+6]=fmaxf(sqrtf(s),1e-8f); }
    if (t == 1) { float s=0; for(int i=0;i<WCOL;++i) s+=wk[i]*wk[i]; ws[OFF_SCAL+7]=fmaxf(sqrtf(s),1e-8f); }
    if (t == 2) { float a=p.rbb[0]; for(int i=0;i<PPD;++i)a+=pps[i]*p.rbw[i]; ws[OFF_SCAL+0]=lrelu(a); }
    if (t == 3) { float a=p.rgb[0]; for(int i=0;i<PPD;++i)a+=pps[i]*p.rgw[i]; ws[OFF_SCAL+1]=sigm(a); }
    if (t == 4) { float a=p.rmb[0]; for(int i=0;i<PPD;++i)a+=pps[i]*p.rmw[i]; ws[OFF_SCAL+2]=fmaxf(a,0.f)+1.f; }
    if (t == 5) { float a=p.wbb[0]; for(int i=0;i<PPD;++i)a+=pps[i]*p.wbw[i]; ws[OFF_SCAL+3]=lrelu(a); }
    if (t == 6) { float a=p.wgb[0]; for(int i=0;i<PPD;++i)a+=pps[i]*p.wgw[i]; ws[OFF_SCAL+4]=sigm(a); }
    if (t == 7) { float a=p.wmb[0]; for(int i=0;i<PPD;++i)a+=pps[i]*p.wmw[i]; ws[OFF_SCAL+5]=fmaxf(a,0.f)+1.f; }
}

// ---- shift distributions: softmax(pp @ Wshift + b) over N=2048 ------------
__global__ void k_shift(Ptrs p, float* ws) {
    const int head = blockIdx.x;               // 0=read 1=write
    const float* W = head ? p.wsw : p.rsw;
    const float* B = head ? p.wsb : p.rsb;
    float* out = ws + (head ? OFF_WSHIFT : OFF_RSHIFT);
    __shared__ float pps[PPD];
    __shared__ float arr[N_MEM];
    __shared__ float red[1024];
    int t = threadIdx.x;                       // 1024 threads
    if (t < PPD) pps[t] = ws[OFF_PP+t];
    __syncthreads();
    for (int o = t; o < N_MEM; o += 1024) {
        float a = B[o];
        for (int i=0;i<PPD;++i) a += pps[i]*W[i*N_MEM+o];
        arr[o] = a;
    }
    __syncthreads();
    red[t] = fmaxf(arr[t], arr[t+1024]); __syncthreads();
    for (int s=512;s>0;s>>=1){ if(t<s) red[t]=fmaxf(red[t],red[t+s]); __syncthreads(); }
    float m = red[0]; __syncthreads();
    float e0 = expf(arr[t]-m), e1 = expf(arr[t+1024]-m);
    red[t] = e0+e1; __syncthreads();
    for (int s=512;s>0;s>>=1){ if(t<s) red[t]+=red[t+s]; __syncthreads(); }
    float inv = 1.f/red[0];
    out[t] = e0*inv; out[t+1024] = e1*inv;
}

// ---- per-row cosine pieces: mem@k (both heads) + row norms (one pass) -----
__global__ void k_cosine(Ptrs p, float* ws) {
    __shared__ float rk[WCOL], wk[WCOL];
    int t = threadIdx.x;                       // 256 threads, grid 8
    if (t < WCOL) { rk[t]=ws[OFF_RKEY+t]; wk[t]=ws[OFF_WKEY+t]; }
    __syncthreads();
    int i = blockIdx.x*256 + t;
    const float* row = p.mem + (size_t)i*WCOL;
    float ar=0.f, aw=0.f, ss=0.f;
    for (int c=0;c<WCOL;++c){ float m=row[c]; ar+=m*rk[c]; aw+=m*wk[c]; ss+=m*m; }
    ws[OFF_RNUM+i]=ar; ws[OFF_WNUM+i]=aw; ws[OFF_ROWDEN+i]=fmaxf(sqrtf(ss),1e-8f);
}

// ---- content softmax + gate interpolation ---------------------------------
__global__ void k_content(Ptrs p, float* ws) {
    const int head = blockIdx.x;
    const float* num  = ws + (head?OFF_WNUM:OFF_RNUM);
    const float* prev = head ? p.prevw : p.prevr;
    float beta = ws[OFF_SCAL + (head?3:0)];
    float gate = ws[OFF_SCAL + (head?4:1)];
    float kn   = ws[OFF_SCAL + 6 + head];
    float* out = ws + (head?OFF_WGW:OFF_RGW);
    __shared__ float arr[N_MEM]; __shared__ float red[1024];
    int t = threadIdx.x;                       // 1024 threads
    for (int i=t;i<N_MEM;i+=1024) arr[i] = beta * (num[i] / (ws[OFF_ROWDEN+i]*kn));
    __syncthreads();
    red[t] = fmaxf(arr[t], arr[t+1024]); __syncthreads();
    for (int s=512;s>0;s>>=1){ if(t<s) red[t]=fmaxf(red[t],red[t+s]); __syncthreads(); }
    float m = red[0]; __syncthreads();
    float e0 = expf(arr[t]-m), e1 = expf(arr[t+1024]-m);
    red[t] = e0+e1; __syncthreads();
    for (int s=512;s>0;s>>=1){ if(t<s) red[t]+=red[t+s]; __syncthreads(); }
    float inv = 1.f/red[0];
    out[t]      = gate*(e0*inv) + (1.f-gate)*prev[t];
    out[t+1024] = gate*(e1*inv) + (1.f-gate)*prev[t+1024];
}

// ---- circular convolution: shifted[i] = sum_j gw[j]*shift[(i-j) mod N] ----
__global__ void k_conv(float* ws) {
    const int head = blockIdx.y;
    const float* gwp = ws + (head?OFF_WGW:OFF_RGW);
    const float* shp = ws + (head?OFF_WSHIFT:OFF_RSHIFT);
    float* out = ws + (head?OFF_WCONV:OFF_RCONV);
    __shared__ float g[N_MEM], s[N_MEM];
    int t = threadIdx.x;                       // 256 threads, grid (8,2)
    for (int i=t;i<N_MEM;i+=256){ g[i]=gwp[i]; s[i]=shp[i]; }
    __syncthreads();
    int i = blockIdx.x*256 + t;
    float acc = 0.f;
    for (int j=0;j<N_MEM;++j) acc += g[j]*s[(i-j)&(N_MEM-1)];
    out[i] = acc;
}

// ---- sharpen: softmax(shifted ** gamma) -----------------------------------
__global__ void k_sharpen(float* ws) {
    const int head = blockIdx.x;
    float gmm = ws[OFF_SCAL + (head?5:2)];
    const float* sp = ws + (head?OFF_WCONV:OFF_RCONV);
    float* out = ws + (head?OFF_WFINAL:OFF_RFINAL);
    __shared__ float arr[N_MEM]; __shared__ float red[1024];
    int t = threadIdx.x;                       // 1024 threads
    for (int i=t;i<N_MEM;i+=1024) {
        float v = sp[i];
        arr[i] = (v > 0.f) ? expf(gmm*logf(v)) : 0.f;  // v**gamma
    }
    __syncthreads();
    red[t] = fmaxf(arr[t], arr[t+1024]); __syncthreads();
    for (int s=512;s>0;s>>=1){ if(t<s) red[t]=fmaxf(red[t],red[t+s]); __syncthreads(); }
    float m = red[0]; __syncthreads();
    float e0 = expf(arr[t]-m), e1 = expf(arr[t+1024]-m);
    red[t] = e0+e1; __syncthreads();
    for (int s=512;s>0;s>>=1){ if(t<s) red[t]+=red[t+s]; __syncthreads(); }
    float inv = 1.f/red[0];
    out[t] = e0*inv; out[t+1024] = e1*inv;
}

// ---- build aug = [x, (r_w^T * memory).flatten()] --------------------------
__global__ void k_aug(Ptrs p, float* ws) {
    int idx = blockIdx.x*256 + threadIdx.x;    // grid 1026 -> exactly 262656
    float* aug = ws + OFF_AUG;
    if (idx < DATA) aug[idx] = p.x[idx];
    else { int t = idx - DATA; aug[idx] = ws[OFF_RFINAL + (t>>7)] * p.mem[t]; }
}

// ---- big GEMV on f32 WMMA: partial[kchunk][j] = sum over 512 k ------------
// A tile 16x4 with only row M=0 live (batch 1). Per ISA 16x4 f32 A layout:
//   VGPR0: lanes0-15 K=0, lanes16-31 K=2 ; VGPR1: K=1 / K=3   (M = lane%16)
// B tile 4x16 mirrored: VGPR0 rows {k,k+2} across lane halves, VGPR1 = +1 row.
// All 8 waves of a block share one k-chunk -> stage aug chunk in LDS once.
// W is 269MB read-once (bigger than 192MB L2) -> non-temporal loads.
__global__ void k_gemv_wmma(const float* __restrict__ aug,
                            const float* __restrict__ W,
                            float* __restrict__ partial) {
    __shared__ float as[KCHUNK];
    const int wid   = threadIdx.x >> 5;        // 8 waves / block
    const int lane  = threadIdx.x & 31;
    const int gwid  = blockIdx.x*8 + wid;      // 0..8207 = 513*16
    const int kch   = gwid >> 4;               // 0..512 (uniform within block)
    const int jt    = gwid & 15;               // 0..15
    const int jb    = jt * 16;
    const int kbase = kch * KCHUNK;
    const int lh    = lane & 15;
    const int hiOff = (lane >= 16) ? 2 : 0;

    for (int i = threadIdx.x; i < KCHUNK; i += 256) as[i] = aug[kbase + i];
    __syncthreads();

    const float* wp = W + (size_t)(kbase + hiOff)*256 + jb + lh;
    const float* ap = as + hiOff;
    v8f c = {};
    for (int kk = 0; kk < KCHUNK; kk += 4) {
        v2f a;
        a.x = (lh == 0) ? ap[0] : 0.0f;        // only M=0 row is live
        a.y = (lh == 0) ? ap[1] : 0.0f;
        v2f b;
        b.x = __builtin_nontemporal_load(wp);        // row k + hiOff
        b.y = __builtin_nontemporal_load(wp + 256);  // row k + hiOff + 1
        c = __builtin_amdgcn_wmma_f32_16x16x4_f32(
                /*neg_a=*/false, a, /*neg_b=*/false, b,
                /*c_mod=*/(short)0, c, /*reuse_a=*/false, /*reuse_b=*/false);
        wp += 1024;                             // +4 rows of 256 floats
        ap += 4;
    }
    // D row M=0 lives in VGPR0 (c[0]) of lanes 0..15, N = lane
    if (lane < 16) partial[(size_t)kch*256 + jb + lane] = c[0];
}

// ---- deterministic reduce + out2 + out3 -> yhat ---------------------------
__global__ void k_final(Ptrs p, const float* ws, float* dout) {
    __shared__ float hb[256], h2s[128];
    int t = threadIdx.x;                       // 512 threads
    if (t < 256) {
        float a = p.o1b[t];
        const float* par = ws + OFF_PARTIAL;
        for (int c=0;c<NKCHUNK;++c) a += par[c*256+t];
        hb[t] = lrelu(a);
    }
    __syncthreads();
    if (t < 128) {
        float a = p.o2b[t];
        for (int i=0;i<256;++i) a += hb[i]*p.o2w[i*128+t];
        h2s[t] = lrelu(a);
    }
    __syncthreads();
    {
        float a = p.o3b[t];
        for (int i=0;i<128;++i) a += h2s[i]*p.o3w[i*OUTD+t];
        dout[t] = sigm(a);
    }
}

// ---- next_mem = mem*(1 - w^T erase) + w^T add (write-once -> NT store) ----
__global__ void k_nextmem(Ptrs p, const float* ws, float* dout) {
    int idx = blockIdx.x*256 + threadIdx.x;    // grid 1024 -> 262144 exact
    int i = idx >> 7, cc = idx & 127;
    float wv = ws[OFF_WFINAL+i];
    float v = p.mem[idx]*(1.f - wv*ws[OFF_ERASE+cc]) + wv*ws[OFF_ADD+cc];
    __builtin_nontemporal_store(v, &dout[OUTD + idx]);
}

extern "C" void kernel_launch(void* const* d_in, const int* in_sizes, int n_in,
                              void* d_out, int out_size, void* d_ws, size_t ws_size,
                              hipStream_t stream) {
    (void)in_sizes; (void)n_in; (void)out_size; (void)ws_size;
    const float* const* F = (const float* const*)d_in;
    Ptrs p;
    p.x=F[0];  p.mem=F[1]; p.prevr=F[2]; p.prevw=F[3];
    p.pp1w=F[4]; p.pp1b=F[5]; p.pp2w=F[6]; p.pp2b=F[7];
    p.rkw=F[8];  p.rkb=F[9];  p.rbw=F[10]; p.rbb=F[11];
    p.rgw=F[12]; p.rgb=F[13]; p.rsw=F[14]; p.rsb=F[15];
    p.rmw=F[16]; p.rmb=F[17];
    p.wkw=F[18]; p.wkb=F[19]; p.wbw=F[20]; p.wbb=F[21];
    p.wgw=F[22]; p.wgb=F[23]; p.wsw=F[24]; p.wsb=F[25];
    p.wmw=F[26]; p.wmb=F[27];
    p.erw=F[28]; p.erb=F[29]; p.adw=F[30]; p.adb=F[31];
    p.o1w=F[32]; p.o1b=F[33]; p.o2w=F[34]; p.o2b=F[35];
    p.o3w=F[36]; p.o3b=F[37];

    float* ws = (float*)d_ws;
    float* out = (float*)d_out;

    k_pp<<<1, 128, 0, stream>>>(p, ws);
    k_heads_small<<<1, 256, 0, stream>>>(p, ws);
    k_shift<<<2, 1024, 0, stream>>>(p, ws);
    k_cosine<<<8, 256, 0, stream>>>(p, ws);
    k_content<<<2, 1024, 0, stream>>>(p, ws);
    k_conv<<<dim3(8,2), 256, 0, stream>>>(ws);
    k_sharpen<<<2, 1024, 0, stream>>>(ws);
    k_aug<<<1026, 256, 0, stream>>>(p, ws);
    k_gemv_wmma<<<1026, 256, 0, stream>>>(ws + OFF_AUG, p.o1w, ws + OFF_PARTIAL);
    k_final<<<1, 512, 0, stream>>>(p, ws, out);
    k_nextmem<<<1024, 256, 0, stream>>>(p, ws, out);
}